// EdgeConv_31044023616094
// MI455X (gfx1250) — compile-verified
//
#include <hip/hip_runtime.h>

#define H 128
#define NN 50000
#define EE 800000
#define MM 100000
#define ELL 800000
#define ALPHA 0.8f
#define BETA 0.2f
#define EPSV 1e-5f

typedef __attribute__((ext_vector_type(2))) float v2f;
typedef __attribute__((ext_vector_type(8))) float v8f;

static inline int cdiv(long long a, long long b) { return (int)((a + b - 1) / b); }

// ---------------- utility ----------------
__global__ void fill_zero_k(float* __restrict__ p, long long n) {
    long long i = (long long)blockIdx.x * blockDim.x + threadIdx.x;
    if (i < n) p[i] = 0.0f;
}

// ---------------- degree / edge normalization ----------------
__global__ void deg_k(const int* __restrict__ row, const float* __restrict__ w,
                      float* __restrict__ deg, int E) {
    int e = blockIdx.x * blockDim.x + threadIdx.x;
    if (e < E) atomicAdd(&deg[row[e]], w[e]);
}

__global__ void dinv_k(float* __restrict__ deg, int n) {
    int i = blockIdx.x * blockDim.x + threadIdx.x;
    if (i < n) {
        float d = deg[i];
        d = (d < 0.5f) ? d + 1.0f : d;
        deg[i] = rsqrtf(d);
    }
}

__global__ void val_k(const int* __restrict__ row, const int* __restrict__ col,
                      const float* __restrict__ w, const float* __restrict__ dinv,
                      float* __restrict__ val, int E) {
    int e = blockIdx.x * blockDim.x + threadIdx.x;
    if (e < E) val[e] = dinv[row[e]] * w[e] * dinv[col[e]];
}

// ---------------- WMMA fp32 GEMM: Y = X @ W^T + b ----------------
// One wave32 computes a 16-row x 128-col block using V_WMMA_F32_16X16X4_F32.
// A fragment (16x4): lane l holds row (l&15), k = k0 + 2*(l>>4) .. +1
// B fragment (4x16): lane l holds col (l&15), k = k0 + 2*(l>>4) .. +1
// C/D (16x16): VGPR r, lanes 0-15 -> (row0+r, n), lanes 16-31 -> (row0+8+r, n)
__global__ __launch_bounds__(256)
void gemm_xwt_bias_k(const float* __restrict__ X, const float* __restrict__ W,
                     const float* __restrict__ bvec, float* __restrict__ Y, int R) {
    int lane = threadIdx.x & 31;
    int wave = (blockIdx.x * (blockDim.x >> 5)) + (threadIdx.x >> 5);
    int row0 = wave * 16;
    if (row0 >= R) return;

    int l15 = lane & 15;
    int lhi = lane >> 4;

    v8f acc[8];
#pragma unroll
    for (int t = 0; t < 8; ++t) {
        float bv = bvec[t * 16 + l15];
#pragma unroll
        for (int r = 0; r < 8; ++r) acc[t][r] = bv;
    }

    const float* __restrict__ xrow = X + (size_t)(row0 + l15) * H;

    for (int k0 = 0; k0 < H; k0 += 4) {
        int ka = k0 + 2 * lhi;
        v2f a;
        a.x = xrow[ka];
        a.y = xrow[ka + 1];
#pragma unroll
        for (int t = 0; t < 8; ++t) {
            int n = t * 16 + l15;
            v2f b;
            b.x = W[(size_t)n * H + ka];
            b.y = W[(size_t)n * H + ka + 1];
            acc[t] = __builtin_amdgcn_wmma_f32_16x16x4_f32(
                false, a, false, b, (short)0, acc[t], false, false);
        }
    }

    int outRow = row0 + lhi * 8;
#pragma unroll
    for (int t = 0; t < 8; ++t) {
#pragma unroll
        for (int r = 0; r < 8; ++r) {
            Y[(size_t)(outRow + r) * H + t * 16 + l15] = acc[t][r];
        }
    }
}

// ---------------- SpMM: out[row] += val * X[col] ----------------
// One wave per edge: 32 lanes x float4 = 128 features.
__global__ void spmm_scatter_k(const int* __restrict__ row, const int* __restrict__ col,
                               const float* __restrict__ val, const float* __restrict__ X,
                               float* __restrict__ out, int E) {
    long long gid = (long long)blockIdx.x * blockDim.x + threadIdx.x;
    long long total = (long long)E * 32;
    if (gid >= total) return;
    int e = (int)(gid >> 5);
    int g = ((int)gid & 31) * 4;
    float v = val[e];
    int c = col[e];
    int r = row[e];
    const float4 x = *(const float4*)(X + (size_t)c * H + g);
    float* o = out + (size_t)r * H + g;
    atomicAdd(o + 0, v * x.x);
    atomicAdd(o + 1, v * x.y);
    atomicAdd(o + 2, v * x.z);
    atomicAdd(o + 3, v * x.w);
}

// ---------------- GraphNorm stats: per-feature sum and sum-of-squares ----------------
__global__ void stats_k(const float* __restrict__ Y, float* __restrict__ accum, int R) {
    int j = threadIdx.x;  // 0..127
    float s1 = 0.0f, s2 = 0.0f;
    for (int r = blockIdx.x; r < R; r += gridDim.x) {
        float y = Y[(size_t)r * H + j];
        s1 += y;
        s2 += y * y;
    }
    atomicAdd(&accum[j], s1);
    atomicAdd(&accum[H + j], s2);
}

// ---------------- GraphNorm apply + ReLU (in place) ----------------
__global__ void norm_relu_k(float* __restrict__ Y, const float* __restrict__ accum,
                            const float* __restrict__ wt, const float* __restrict__ bs,
                            const float* __restrict__ ms, int R) {
    long long gid = (long long)blockIdx.x * blockDim.x + threadIdx.x;
    if (gid >= (long long)R * H) return;
    int j = (int)(gid & (H - 1));
    float invR = 1.0f / (float)R;
    float mean = accum[j] * invR;
    float m2 = accum[H + j] * invR;
    float mu = ms[j] * mean;
    float var = m2 - 2.0f * mu * mean + mu * mu;  // E[(y-mu)^2]
    float sub = Y[gid] - mu;
    float o = wt[j] * sub * rsqrtf(var + EPSV) + bs[j];
    Y[gid] = fmaxf(o, 0.0f);
}

// ---------------- Mixing ----------------
__global__ void local_mix_k(const float* __restrict__ local_h, const float* __restrict__ base_h,
                            const int* __restrict__ c2o, float* __restrict__ local_mixed,
                            float* __restrict__ summ) {
    long long gid = (long long)blockIdx.x * blockDim.x + threadIdx.x;
    if (gid >= (long long)MM * H) return;
    int i = (int)(gid >> 7);
    int j = (int)(gid & (H - 1));
    int o = c2o[i];
    float lm = ALPHA * local_h[gid] + BETA * base_h[(size_t)o * H + j];
    local_mixed[gid] = lm;
    atomicAdd(&summ[(size_t)o * H + j], lm);
}

__global__ void cnt_k(const int* __restrict__ c2o, float* __restrict__ cnt) {
    int i = blockIdx.x * blockDim.x + threadIdx.x;
    if (i < MM) atomicAdd(&cnt[c2o[i]], 1.0f);
}

__global__ void base_mix_k(const float* __restrict__ base_h, const float* __restrict__ summ,
                           const float* __restrict__ cnt, float* __restrict__ base_mixed) {
    long long gid = (long long)blockIdx.x * blockDim.x + threadIdx.x;
    if (gid >= (long long)NN * H) return;
    int i = (int)(gid >> 7);
    float c = fmaxf(cnt[i], 1.0f);
    base_mixed[gid] = ALPHA * base_h[gid] + BETA * (summ[gid] / c);
}

// ---------------- host-side orchestration ----------------
extern "C" void kernel_launch(void* const* d_in, const int* in_sizes, int n_in,
                              void* d_out, int out_size, void* d_ws, size_t ws_size,
                              hipStream_t stream) {
    const float* base_x = (const float*)d_in[0];
    const int*   bei    = (const int*)d_in[1];      // [2,E]
    const float* bew    = (const float*)d_in[2];
    const float* local_x = (const float*)d_in[3];
    const int*   lai    = (const int*)d_in[4];      // [2,EL]
    const float* lav    = (const float*)d_in[5];
    const int*   c2o    = (const int*)d_in[6];
    const float* base_W = (const float*)d_in[7];
    const float* base_b = (const float*)d_in[8];
    const float* local_W = (const float*)d_in[9];
    const float* local_b = (const float*)d_in[10];
    const float* bgn_w = (const float*)d_in[11];
    const float* bgn_b = (const float*)d_in[12];
    const float* bgn_m = (const float*)d_in[13];
    const float* lgn_w = (const float*)d_in[14];
    const float* lgn_b = (const float*)d_in[15];
    const float* lgn_m = (const float*)d_in[16];

    const int* brow = bei;           // row = edge_index[0]
    const int* bcol = bei + EE;      // col = edge_index[1]
    const int* lrow = lai;
    const int* lcol = lai + ELL;

    // workspace layout (floats)
    float* ws = (float*)d_ws;
    float* deg   = ws;                         // N
    float* val   = deg + NN;                   // E
    float* bx    = val + EE;                   // N*H   (reused as summ later)
    float* yb    = bx + (size_t)NN * H;        // N*H   (spmm base -> base_h)
    float* lx    = yb + (size_t)NN * H;        // M*H
    float* yl    = lx + (size_t)MM * H;        // M*H   (spmm local -> local_h)
    float* cnt   = yl + (size_t)MM * H;        // N
    float* accB  = cnt + NN;                   // 2*H
    float* accL  = accB + 2 * H;               // 2*H

    float* base_mixed  = (float*)d_out;                  // [N,H]
    float* local_mixed = (float*)d_out + (size_t)NN * H; // [M,H]

    const int T = 256;

    // zero-init scratch used as accumulators
    fill_zero_k<<<cdiv(NN, T), T, 0, stream>>>(deg, NN);
    fill_zero_k<<<cdiv((long long)NN * H, T), T, 0, stream>>>(yb, (long long)NN * H);
    fill_zero_k<<<cdiv((long long)MM * H, T), T, 0, stream>>>(yl, (long long)MM * H);
    fill_zero_k<<<cdiv(NN, T), T, 0, stream>>>(cnt, NN);
    fill_zero_k<<<1, T, 0, stream>>>(accB, 2 * H);
    fill_zero_k<<<1, T, 0, stream>>>(accL, 2 * H);

    // GCN edge normalization
    deg_k<<<cdiv(EE, T), T, 0, stream>>>(brow, bew, deg, EE);
    dinv_k<<<cdiv(NN, T), T, 0, stream>>>(deg, NN);
    val_k<<<cdiv(EE, T), T, 0, stream>>>(brow, bcol, bew, deg, val, EE);

    // base path: GEMM (WMMA fp32) -> SpMM -> GraphNorm+ReLU
    {
        int waves = NN / 16;                    // 3125
        int blocks = cdiv(waves, T / 32);
        gemm_xwt_bias_k<<<blocks, T, 0, stream>>>(base_x, base_W, base_b, bx, NN);
    }
    spmm_scatter_k<<<cdiv((long long)EE * 32, T), T, 0, stream>>>(brow, bcol, val, bx, yb, EE);
    stats_k<<<1024, H, 0, stream>>>(yb, accB, NN);
    norm_relu_k<<<cdiv((long long)NN * H, T), T, 0, stream>>>(yb, accB, bgn_w, bgn_b, bgn_m, NN);

    // local path
    {
        int waves = MM / 16;                    // 6250
        int blocks = cdiv(waves, T / 32);
        gemm_xwt_bias_k<<<blocks, T, 0, stream>>>(local_x, local_W, local_b, lx, MM);
    }
    spmm_scatter_k<<<cdiv((long long)ELL * 32, T), T, 0, stream>>>(lrow, lcol, lav, lx, yl, ELL);
    stats_k<<<1024, H, 0, stream>>>(yl, accL, MM);
    norm_relu_k<<<cdiv((long long)MM * H, T), T, 0, stream>>>(yl, accL, lgn_w, lgn_b, lgn_m, MM);

    // mixing: reuse bx as summ accumulator
    fill_zero_k<<<cdiv((long long)NN * H, T), T, 0, stream>>>(bx, (long long)NN * H);
    local_mix_k<<<cdiv((long long)MM * H, T), T, 0, stream>>>(yl, yb, c2o, local_mixed, bx);
    cnt_k<<<cdiv(MM, T), T, 0, stream>>>(c2o, cnt);
    base_mix_k<<<cdiv((long long)NN * H, T), T, 0, stream>>>(yb, bx, cnt, base_mixed);
}